// Group_maskpoint_51247549775876
// MI455X (gfx1250) — compile-verified
//
#include <hip/hip_runtime.h>

typedef __attribute__((ext_vector_type(2))) float v2f;
typedef __attribute__((ext_vector_type(8))) float v8f;

#define BATCH   16
#define NPTS    8192
#define NGROUP  512
#define GSIZE   32
#define TC      16          // centers per KNN block (one WMMA M-tile)
#define CHUNK   512         // points per streamed chunk
#define CPAD    (CHUNK + 4) // 8-row offset = 32 banks -> conflict-free WMMA stores
#define SPAD    (CHUNK + 16)// 2-row offset = 32 banks -> conflict-free B-operand reads

// ---------------- async global->LDS staging (gfx1250 ASYNCcnt path) ----------
#if __has_builtin(__builtin_amdgcn_global_load_async_to_lds_b32) && \
    __has_builtin(__builtin_amdgcn_s_wait_asynccnt)
#define HAVE_ASYNC_LDS 1
#else
#define HAVE_ASYNC_LDS 0
#endif

__device__ __forceinline__ void async_copy_f32(const float* g, float* l) {
#if HAVE_ASYNC_LDS
    __builtin_amdgcn_global_load_async_to_lds_b32(
        (__attribute__((address_space(1))) int*)g,
        (__attribute__((address_space(3))) int*)l, 0, 0);
#else
    *l = *g;
#endif
}
__device__ __forceinline__ void async_join() {
#if HAVE_ASYNC_LDS
    __builtin_amdgcn_s_wait_asynccnt(0);
#endif
}

// ---------------- wave32 bitonic compare-exchange on (dist, idx) pairs -------
__device__ __forceinline__ void cmpx(float& d, int& i, int j, bool up, int lane) {
    float od = __shfl_xor(d, j, 32);
    int   oi = __shfl_xor(i, j, 32);
    bool otherLess = (od < d) || (od == d && oi < i);   // lexicographic (d, idx)
    bool upper = (lane & j) != 0;
    bool take = up ? (upper ? !otherLess : otherLess)
                   : (upper ? otherLess : !otherLess);
    if (take) { d = od; i = oi; }
}

// =============================================================================
// Kernel 1: farthest point sampling. One workgroup per batch; cloud in LDS,
// running min-distance in registers; 511 rounds of block-wide argmax.
// =============================================================================
__global__ __launch_bounds__(1024) void fps_kernel(const float* __restrict__ xyz,
                                                   float* __restrict__ center) {
    __shared__ float sx[NPTS], sy[NPTS], sz[NPTS];   // 96 KB of the 320 KB WGP LDS
    __shared__ float rv[32];
    __shared__ int   ri[32];
    __shared__ int   sbest;

    const int b    = blockIdx.x;
    const int tid  = threadIdx.x;
    const int lane = tid & 31;
    const int wave = tid >> 5;
    const float* xb = xyz + (size_t)b * NPTS * 3;

    float dist[8];
    #pragma unroll
    for (int j = 0; j < 8; ++j) {
        int p = tid + j * 1024;                       // strided: conflict-free LDS
        async_copy_f32(xb + (size_t)p * 3 + 0, &sx[p]);
        async_copy_f32(xb + (size_t)p * 3 + 1, &sy[p]);
        async_copy_f32(xb + (size_t)p * 3 + 2, &sz[p]);
        dist[j] = 1e10f;
    }
    async_join();
    __syncthreads();

    float* cb = center + (size_t)b * NGROUP * 3;
    if (tid == 0) { cb[0] = sx[0]; cb[1] = sy[0]; cb[2] = sz[0]; }  // idx[0] = 0

    int last = 0;
    for (int it = 1; it < NGROUP; ++it) {
        float lx = sx[last], ly = sy[last], lz = sz[last];  // LDS broadcast
        float bv = -1.0f; int bi = 0x7fffffff;
        #pragma unroll
        for (int j = 0; j < 8; ++j) {
            int p = tid + j * 1024;
            float dx = sx[p] - lx, dy = sy[p] - ly, dz = sz[p] - lz;
            float d  = fmaf(dx, dx, fmaf(dy, dy, dz * dz));
            float nd = fminf(dist[j], d);
            dist[j] = nd;
            if (nd > bv || (nd == bv && p < bi)) { bv = nd; bi = p; }
        }
        // wave32 argmax (first-index tie-break == jnp.argmax)
        #pragma unroll
        for (int off = 16; off > 0; off >>= 1) {
            float ov = __shfl_xor(bv, off, 32);
            int   oi = __shfl_xor(bi, off, 32);
            if (ov > bv || (ov == bv && oi < bi)) { bv = ov; bi = oi; }
        }
        if (lane == 0) { rv[wave] = bv; ri[wave] = bi; }
        __syncthreads();
        if (wave == 0) {
            float v = rv[lane]; int i = ri[lane];
            #pragma unroll
            for (int off = 16; off > 0; off >>= 1) {
                float ov = __shfl_xor(v, off, 32);
                int   oi = __shfl_xor(i, off, 32);
                if (ov > v || (ov == v && oi < i)) { v = ov; i = oi; }
            }
            if (lane == 0) sbest = i;
        }
        __syncthreads();
        last = sbest;
        if (tid == 0) {
            cb[it * 3 + 0] = sx[last];
            cb[it * 3 + 1] = sy[last];
            cb[it * 3 + 2] = sz[last];
        }
    }
}

// =============================================================================
// Kernel 2: KNN + gather. Block = 16 waves handles 16 centers of one batch.
// d2 tiles via V_WMMA_F32_16X16X4_F32 with branchless operand addressing
// (5-row LDS tables: x,y,z,zero,norm); per-center top-32 kept distributed
// across one wave (K == wave width) with ballot-gated bitonic merges.
// =============================================================================
__global__ __launch_bounds__(512) void knn_kernel(const float* __restrict__ xyz,
                                                  const float* __restrict__ center,
                                                  float* __restrict__ neigh) {
    __shared__ float d2s[TC][CPAD];     // 33.0 KB
    __shared__ float sp[5 * SPAD];      // point rows: x,y,z,zero,norm (10.5 KB)
    __shared__ float sc[5 * TC];        // center rows: x,y,z,zero,norm

    const int b    = blockIdx.y;
    const int tile = blockIdx.x;        // 0..31 (512/16 tiles)
    const int tid  = threadIdx.x;
    const int lane = tid & 31;
    const int wave = tid >> 5;          // 0..15 == center row
    const float* xb    = xyz + (size_t)b * NPTS * 3;
    const float* cbase = center + ((size_t)b * NGROUP + (size_t)tile * TC) * 3;

    if (tid < TC) {
        float x = cbase[tid * 3 + 0], y = cbase[tid * 3 + 1], z = cbase[tid * 3 + 2];
        sc[0 * TC + tid] = x;
        sc[1 * TC + tid] = y;
        sc[2 * TC + tid] = z;
        sc[3 * TC + tid] = 0.0f;                      // K=3 zero pad row
        sc[4 * TC + tid] = fmaf(x, x, fmaf(y, y, z * z));
    }
    sp[3 * SPAD + tid] = 0.0f;                        // K=3 zero pad row (once)

    float curd = 1e30f;                               // running top-32, lane = rank
    int   curi = 0x7fffffff;

    for (int cb0 = 0; cb0 < NPTS; cb0 += CHUNK) {
        __syncthreads();                              // prev-phase readers done
        {   // stage chunk points into LDS via async global->LDS copies
            int p = cb0 + tid;
            async_copy_f32(xb + (size_t)p * 3 + 0, &sp[0 * SPAD + tid]);
            async_copy_f32(xb + (size_t)p * 3 + 1, &sp[1 * SPAD + tid]);
            async_copy_f32(xb + (size_t)p * 3 + 2, &sp[2 * SPAD + tid]);
            if (cb0 + CHUNK < NPTS)                   // global_prefetch next chunk
                __builtin_prefetch(xb + (size_t)(cb0 + CHUNK + tid) * 3, 0, 0);
        }
        async_join();
        __syncthreads();
        {   // point norms
            float x = sp[0 * SPAD + tid], y = sp[1 * SPAD + tid], z = sp[2 * SPAD + tid];
            sp[4 * SPAD + tid] = fmaf(x, x, fmaf(y, y, z * z));
        }
        __syncthreads();

        // ---- WMMA phase: each wave computes 2 tiles of 16 points ----
        #pragma unroll
        for (int t = 0; t < 2; ++t) {
            int ptl = (wave * 2 + t) * 16;            // chunk-local col base
            int m   = lane & 15;
            int n   = ptl + m;
            int ka  = (lane >> 4) << 1;               // 0 (lanes 0-15) / 2 (16-31)
            // A = centers 16x4: VGPR0 = K0|K2, VGPR1 = K1|K3 (row 3 is zeros)
            v2f a, bb;
            a[0]  = sc[ka * TC + m];
            a[1]  = sc[(ka + 1) * TC + m];
            // B = points 4x16: VGPR0 = K0|K2, VGPR1 = K1|K3 (row 3 is zeros)
            bb[0] = sp[ka * SPAD + n];
            bb[1] = sp[(ka + 1) * SPAD + n];
            v8f c = {};
            v8f d = __builtin_amdgcn_wmma_f32_16x16x4_f32(
                false, a, false, bb, (short)0, c, false, false);
            float pn = sp[4 * SPAD + n];
            int rbase = (lane >> 4) << 3;             // C layout: M = r + 8*hi
            #pragma unroll
            for (int r = 0; r < 8; ++r) {
                int row = rbase + r;
                d2s[row][n] = fmaf(-2.0f, d[r], sc[4 * TC + row] + pn);
            }
        }
        __syncthreads();

        // ---- top-k phase: wave handles its center row ----
        #pragma unroll 1
        for (int grp = 0; grp < CHUNK / 32; ++grp) {
            int   col = grp * 32 + lane;
            float nd  = d2s[wave][col];
            int   ni  = cb0 + col;
            float worst = __shfl(curd, 31, 32);
            // incoming idx always > stored idx, so strict < is the exact test
            if (__ballot(nd < worst)) {
                // bitonic sort incoming descending by (d, idx)
                #pragma unroll
                for (int k = 2; k <= 32; k <<= 1)
                    #pragma unroll
                    for (int j = k >> 1; j > 0; j >>= 1)
                        cmpx(nd, ni, j, (lane & k) != 0, lane);
                // elementwise lex-min vs ascending current list -> bitonic
                bool incLess = (nd < curd) || (nd == curd && ni < curi);
                if (incLess) { curd = nd; curi = ni; }
                // bitonic merge back to ascending
                #pragma unroll
                for (int j = 16; j > 0; j >>= 1)
                    cmpx(curd, curi, j, true, lane);
            }
        }
    }

    // lane l holds the l-th nearest neighbor of center `wave`: gather + recenter
    {
        int gi = tile * TC + wave;
        int p  = curi;
        float x = xb[p * 3 + 0] - sc[0 * TC + wave];
        float y = xb[p * 3 + 1] - sc[1 * TC + wave];
        float z = xb[p * 3 + 2] - sc[2 * TC + wave];
        float* o = neigh + (((size_t)b * NGROUP + gi) * GSIZE + lane) * 3;
        o[0] = x; o[1] = y; o[2] = z;
    }
}

// =============================================================================
extern "C" void kernel_launch(void* const* d_in, const int* in_sizes, int n_in,
                              void* d_out, int out_size, void* d_ws, size_t ws_size,
                              hipStream_t stream) {
    (void)in_sizes; (void)n_in; (void)d_ws; (void)ws_size; (void)out_size;
    const float* xyz = (const float*)d_in[0];
    float* out    = (float*)d_out;
    float* neigh  = out;                                          // [B,G,32,3]
    float* center = out + (size_t)BATCH * NGROUP * GSIZE * 3;     // [B,G,3]

    fps_kernel<<<dim3(BATCH), dim3(1024), 0, stream>>>(xyz, center);
    knn_kernel<<<dim3(NGROUP / TC, BATCH), dim3(512), 0, stream>>>(xyz, center, neigh);
}